// MoELora_85744727098033
// MI455X (gfx1250) — compile-verified
//
#include <hip/hip_runtime.h>
#include <hip/hip_bf16.h>

typedef __attribute__((ext_vector_type(2))) float v2f;
typedef __attribute__((ext_vector_type(8))) float v8f;

#define B_   4096
#define FF   16
#define HH   1280
#define NN   8
#define RR   64
#define XS_STRIDE 1284   // 1280 + 4 pad -> A-frag LDS reads hit distinct banks
#define MS_STRIDE 68     // 64 + 4 pad

// ---------------------------------------------------------------------------
// Pre-kernel: L2 norms of the 8 gate_w rows (each 16*H = 20480 f32).
// One block, wave n reduces row n.
// ---------------------------------------------------------------------------
__global__ __launch_bounds__(256) void gnorm_kernel(const float* __restrict__ gw,
                                                    float* __restrict__ gn) {
    const int w = threadIdx.x >> 5, lane = threadIdx.x & 31;
    const float* row = gw + (size_t)w * (FF * HH);
    float s = 0.f;
    for (int c = lane * 4; c < FF * HH; c += 32 * 4) {
        float4 v = *(const float4*)&row[c];
        s += v.x * v.x + v.y * v.y + v.z * v.z + v.w * v.w;
    }
#pragma unroll
    for (int o = 16; o > 0; o >>= 1) s += __shfl_xor(s, o, 32);
    if (lane == 0) gn[w] = sqrtf(s);
}

// ---------------------------------------------------------------------------
// Fused gate + single-expert LoRA. One block per batch row.
// ---------------------------------------------------------------------------
__global__ __launch_bounds__(256) void moe_lora_kernel(
    const float* __restrict__ x,      // [B,16,H]
    const float* __restrict__ u,      // [B,N]
    const float* __restrict__ gate_w, // [N,16H]
    const float* __restrict__ sigma,  // [1]
    const float* __restrict__ down_w, // [N,R,H]
    const float* __restrict__ up_w,   // [N,H,R]
    const float* __restrict__ gnorm,  // [N] (workspace)
    float* __restrict__ out)          // [B,16,H]
{
    __shared__ __align__(16) float xs[FF * XS_STRIDE];   // x[b], padded rows
    __shared__ __align__(16) float mids[FF * MS_STRIDE]; // 16x64 mid, padded
    __shared__ float sdot[NN];
    __shared__ float swsum[NN];
    __shared__ int   snsel;

    const int b    = blockIdx.x;
    const int tid  = threadIdx.x;
    const int w    = tid >> 5;
    const int lane = tid & 31;

    // ---- 1) x[b] -> LDS (float4, coalesced) + sum of squares ----
    const float4* xg = (const float4*)(x + (size_t)b * (FF * HH));
    float ss = 0.f;
    for (int e = tid; e < (FF * HH) / 4; e += 256) {
        int row = e / (HH / 4);
        int c4  = e - row * (HH / 4);
        float4 v = xg[e];
        *(float4*)&xs[row * XS_STRIDE + c4 * 4] = v;
        ss += v.x * v.x + v.y * v.y + v.z * v.z + v.w * v.w;
    }
#pragma unroll
    for (int o = 16; o > 0; o >>= 1) ss += __shfl_xor(ss, o, 32);
    if (lane == 0) swsum[w] = ss;
    __syncthreads();

    // ---- 2) gate dot: wave n computes <x, gate_w[n]> (gate_w is L2-resident) ----
    {
        const float* grow = gate_w + (size_t)w * (FF * HH);
        float d = 0.f;
        for (int f = 0; f < FF; ++f) {
            const float* xr = &xs[f * XS_STRIDE];
            const float* gr = &grow[(size_t)f * HH];
            for (int c = lane * 4; c < HH; c += 32 * 4) {
                float4 a = *(const float4*)&xr[c];
                float4 g = *(const float4*)&gr[c];
                d += a.x * g.x + a.y * g.y + a.z * g.z + a.w * g.w;
            }
        }
#pragma unroll
        for (int o = 16; o > 0; o >>= 1) d += __shfl_xor(d, o, 32);
        if (lane == 0) sdot[w] = d;
    }
    __syncthreads();

    // ---- 3) gumbel + argmax (softmax is monotone; hard one-hot forward) ----
    if (tid == 0) {
        float ssum = 0.f;
        for (int n = 0; n < NN; ++n) ssum += swsum[n];
        float nx  = fmaxf(sqrtf(ssum), 1e-12f);
        float sig = sigma[0];
        int best = 0; float bk = -3.402823466e38f;
        for (int n = 0; n < NN; ++n) {
            float ng    = fmaxf(gnorm[n], 1e-12f);
            float logit = sig * sdot[n] / (nx * ng);
            float uu    = u[(size_t)b * NN + n];
            float g     = -logf(-logf(uu + 1e-12f) + 1e-12f);
            float key   = logit + g;                    // TAU == 1
            if (key > bk) { bk = key; best = n; }       // first-max like argmax
        }
        snsel = best;
    }
    __syncthreads();
    const int nsel = snsel;

    // Fragment geometry (f32 WMMA 16x16x4): lanes 0-15 K={0,1}, 16-31 K={2,3}
    const int m  = lane & 15;
    const int kx = (lane >> 4) * 2;

    // ---- 4) down-proj: mid(16x64) = x(16x1280) @ down_w[n]^T(1280x64) ----
    // 8 waves = 4 N-tiles x 2 K-halves; halves summed through LDS.
    {
        const int j  = w & 3;      // N-tile (r block)
        const int kh = w >> 2;     // K half
        const float* dw   = down_w + (size_t)nsel * (RR * HH);      // [64][1280]
        const float* brow = dw + (size_t)(j * 16 + m) * HH + kx;    // B[K,N]=dw[r][c]
        const float* arow = &xs[m * XS_STRIDE + kx];
        const int kbeg = kh * (HH / 2);
        v8f acc = {};
#pragma unroll 4
        for (int k0 = 0; k0 < HH / 2; k0 += 4) {
            v2f a  = *(const v2f*)&arow[kbeg + k0];
            v2f bb = *(const v2f*)&brow[kbeg + k0];
            acc = __builtin_amdgcn_wmma_f32_16x16x4_f32(
                false, a, false, bb, (short)0, acc, false, false);
        }
        const int col = j * 16 + m;
        const int rhi = 8 * (lane >> 4);
        if (kh == 0) {
#pragma unroll
            for (int i = 0; i < 8; ++i)
                mids[(i + rhi) * MS_STRIDE + col] = acc[i];
        }
        __syncthreads();
        if (kh == 1) {
#pragma unroll
            for (int i = 0; i < 8; ++i)
                mids[(i + rhi) * MS_STRIDE + col] += acc[i];
        }
        __syncthreads();
    }

    // ---- 5) up-proj: out(16x1280) = mid(16x64) @ up_w[n]^T(64x1280) ----
    // 80 N-tiles, 10 per wave. A-fragments of mid cached in registers.
    {
        v2f afr[16];
#pragma unroll
        for (int kk = 0; kk < 16; ++kk)
            afr[kk] = *(const v2f*)&mids[m * MS_STRIDE + kk * 4 + kx];

        const float* uw = up_w + (size_t)nsel * (HH * RR);   // [1280][64]
        float* ob = out + (size_t)b * (FF * HH);
        const int rhi = 8 * (lane >> 4);
#pragma unroll 1
        for (int t = 0; t < 10; ++t) {
            const int h = (w * 10 + t) * 16 + m;             // out column
            const float* bp = uw + (size_t)h * RR + kx;      // B[K,N]=up[h][r]
            v8f acc = {};
#pragma unroll
            for (int kk = 0; kk < 16; ++kk) {
                v2f bb = *(const v2f*)&bp[kk * 4];
                acc = __builtin_amdgcn_wmma_f32_16x16x4_f32(
                    false, afr[kk], false, bb, (short)0, acc, false, false);
            }
#pragma unroll
            for (int i = 0; i < 8; ++i)
                ob[(size_t)(i + rhi) * HH + h] = acc[i];
        }
    }
}

// ---------------------------------------------------------------------------
extern "C" void kernel_launch(void* const* d_in, const int* in_sizes, int n_in,
                              void* d_out, int out_size, void* d_ws, size_t ws_size,
                              hipStream_t stream) {
    (void)in_sizes; (void)n_in; (void)out_size; (void)ws_size;
    const float* x      = (const float*)d_in[0];
    const float* u      = (const float*)d_in[1];
    const float* gate_w = (const float*)d_in[2];
    const float* sigma  = (const float*)d_in[3];
    const float* down_w = (const float*)d_in[4];
    const float* up_w   = (const float*)d_in[5];
    float* out   = (float*)d_out;
    float* gnorm = (float*)d_ws;   // 8 floats of scratch

    gnorm_kernel<<<1, 256, 0, stream>>>(gate_w, gnorm);
    moe_lora_kernel<<<B_, 256, 0, stream>>>(x, u, gate_w, sigma, down_w, up_w,
                                            gnorm, out);
}